// GCN_69595650064676
// MI455X (gfx1250) — compile-verified
//
#include <hip/hip_runtime.h>
#include <hip/hip_bf16.h>

typedef float v2f __attribute__((ext_vector_type(2)));
typedef float v8f __attribute__((ext_vector_type(8)));

#define D 64

// ---------------- degree / norm precompute ----------------

__global__ __launch_bounds__(256) void k_degree(const int* __restrict__ col,
                                                const float* __restrict__ ew,
                                                float* __restrict__ deg,
                                                int E, int N) {
    int t = blockIdx.x * blockDim.x + threadIdx.x;
    if (t < E) {
        unsafeAtomicAdd(&deg[col[t]], ew[t]);
    } else if (t < E + N) {
        unsafeAtomicAdd(&deg[t - E], 1.0f);   // self loop, fill_value=1.0
    }
}

__global__ __launch_bounds__(256) void k_dinv(const float* __restrict__ deg,
                                              float* __restrict__ dinv, int N) {
    int n = blockIdx.x * blockDim.x + threadIdx.x;
    if (n < N) {
        float d = deg[n];
        dinv[n] = (d > 0.0f) ? rsqrtf(d) : 0.0f;
    }
}

__global__ __launch_bounds__(256) void k_norm(const int* __restrict__ row,
                                              const int* __restrict__ col,
                                              const float* __restrict__ ew,
                                              const float* __restrict__ dinv,
                                              float* __restrict__ norm, int E) {
    int e = blockIdx.x * blockDim.x + threadIdx.x;
    if (e < E) norm[e] = dinv[row[e]] * ew[e] * dinv[col[e]];
}

// ---------------- dense transform: H[N,64] @ W[64,64] via fp32 WMMA ----------------
// Block: 256 threads = 8 waves. W staged in LDS. Each wave -> one 16x16 D tile.
// 2 row-strips x 4 col-tiles per block => 32 rows / block.

__global__ __launch_bounds__(256) void k_gemm64_wmma(const float* __restrict__ H,
                                                     const float* __restrict__ W,
                                                     float* __restrict__ out,
                                                     int nrows) {
    __shared__ float lw[D * D];
    int tid = threadIdx.x;

    // cooperative load of W (4096 floats) into LDS as float4
    const float4* w4 = (const float4*)W;
    float4* l4 = (float4*)lw;
#pragma unroll
    for (int i = 0; i < 4; ++i) l4[tid + i * 256] = w4[tid + i * 256];
    __syncthreads();

    int wave = tid >> 5;
    int lane = tid & 31;
    int strip = wave >> 2;          // 0..1
    int ctile = wave & 3;           // 0..3
    int row0 = blockIdx.x * 32 + strip * 16;
    int r    = lane & 15;           // A: row within tile / B,D: col within tile
    int koff = (lane >> 4) * 2;     // K sub-pair select
    int colbase = ctile * 16;

    int hr = row0 + r;
    if (hr >= nrows) hr = nrows - 1;            // clamp (grid divides exactly; safety)
    const float* hrow = H + (size_t)hr * D;

    v8f acc = {};
#pragma unroll
    for (int k = 0; k < D; k += 4) {
        v2f a, b;
        a.x = hrow[k + koff];
        a.y = hrow[k + koff + 1];
        b.x = lw[(k + koff)     * D + colbase + r];
        b.y = lw[(k + koff + 1) * D + colbase + r];
        acc = __builtin_amdgcn_wmma_f32_16x16x4_f32(
                  /*neg_a=*/false, a, /*neg_b=*/false, b,
                  /*c_mod=*/(short)0, acc, /*reuse_a=*/false, /*reuse_b=*/false);
    }

    // D tile store: lane<16 -> rows row0+v, lane>=16 -> rows row0+8+v, col = colbase+r
    int mbase = row0 + ((lane < 16) ? 0 : 8);
    float* dst = out + (size_t)mbase * D + colbase + r;
    if (row0 + 16 <= nrows) {
        // full tile: unconditional stores (common path, single branch)
#pragma unroll
        for (int v = 0; v < 8; ++v) dst[(size_t)v * D] = acc[v];
    } else {
#pragma unroll
        for (int v = 0; v < 8; ++v) {
            if (mbase + v < nrows) dst[(size_t)v * D] = acc[v];
        }
    }
}

// ---------------- accumulator init: out = bias (+ residual) ----------------

__global__ __launch_bounds__(256) void k_init_bias(float* __restrict__ out,
                                                   const float* __restrict__ bias,
                                                   const float* __restrict__ res,
                                                   int total) {
    int t = blockIdx.x * blockDim.x + threadIdx.x;
    if (t >= total) return;
    float v = bias[t & (D - 1)];
    if (res) v += res[t];
    out[t] = v;
}

// ---------------- edge scatter: out[col] += norm * hw[row], 64 feats ----------------
// One wave per (virtual) edge; lane handles 2 features. e>=E are self loops.

__global__ __launch_bounds__(256) void k_scatter64(const int* __restrict__ row,
                                                   const int* __restrict__ col,
                                                   const float* __restrict__ norm,
                                                   const float* __restrict__ dinv,
                                                   const float* __restrict__ hw,
                                                   float* __restrict__ out,
                                                   int E, int N) {
    int t = blockIdx.x * blockDim.x + threadIdx.x;
    int e = t >> 5;
    int lane = t & 31;
    if (e >= E + N) return;
    int r, c; float w;
    if (e < E) {
        r = row[e]; c = col[e]; w = norm[e];
    } else {
        r = c = e - E;
        float d = dinv[r];
        w = d * d;                  // dinv[i] * 1.0 * dinv[i]
    }
    const float2 hv = *(const float2*)(hw + (size_t)r * D + lane * 2);
    float* dst = out + (size_t)c * D + lane * 2;
    unsafeAtomicAdd(dst,     w * hv.x);
    unsafeAtomicAdd(dst + 1, w * hv.y);
}

// ---------------- ReLU in place ----------------

__global__ __launch_bounds__(256) void k_relu(float* __restrict__ x, int total) {
    int t = blockIdx.x * blockDim.x + threadIdx.x;
    if (t < total) x[t] = fmaxf(x[t], 0.0f);
}

// ---------------- final projection: hw1[n] = dot(H[n,:], Wf[:,0]) ----------------

__global__ __launch_bounds__(256) void k_dot64(const float* __restrict__ H,
                                               const float* __restrict__ Wf,
                                               float* __restrict__ out, int N) {
    int n = blockIdx.x * blockDim.x + threadIdx.x;
    if (n >= N) return;
    const float4* h4 = (const float4*)(H + (size_t)n * D);
    const float4* w4 = (const float4*)Wf;
    float s = 0.0f;
#pragma unroll
    for (int i = 0; i < 16; ++i) {
        float4 h = h4[i]; float4 w = w4[i];
        s += h.x * w.x + h.y * w.y + h.z * w.z + h.w * w.w;
    }
    out[n] = s;
}

__global__ __launch_bounds__(256) void k_init_scalar(float* __restrict__ out,
                                                     const float* __restrict__ bf,
                                                     int N) {
    int n = blockIdx.x * blockDim.x + threadIdx.x;
    if (n < N) out[n] = bf[0];
}

__global__ __launch_bounds__(256) void k_scatter1(const int* __restrict__ row,
                                                  const int* __restrict__ col,
                                                  const float* __restrict__ norm,
                                                  const float* __restrict__ dinv,
                                                  const float* __restrict__ hw1,
                                                  float* __restrict__ out,
                                                  int E, int N) {
    int e = blockIdx.x * blockDim.x + threadIdx.x;
    if (e >= E + N) return;
    int r, c; float w;
    if (e < E) { r = row[e]; c = col[e]; w = norm[e]; }
    else       { r = c = e - E; float d = dinv[r]; w = d * d; }
    unsafeAtomicAdd(&out[c], w * hw1[r]);
}

// ---------------- host side ----------------

static inline int cdiv(long long a, long long b) { return (int)((a + b - 1) / b); }

extern "C" void kernel_launch(void* const* d_in, const int* in_sizes, int n_in,
                              void* d_out, int out_size, void* d_ws, size_t ws_size,
                              hipStream_t stream) {
    const float* x    = (const float*)d_in[0];
    const int*   ei   = (const int*)d_in[1];      // [2, E]
    const float* ew   = (const float*)d_in[2];
    const float* W_in = (const float*)d_in[4];
    const float* b_in = (const float*)d_in[5];
    const float* W1   = (const float*)d_in[6];
    const float* b1   = (const float*)d_in[7];
    const float* W2   = (const float*)d_in[8];
    const float* b2   = (const float*)d_in[9];
    const float* Wf   = (const float*)d_in[10];
    const float* bf   = (const float*)d_in[11];

    const int N = in_sizes[0] / D;
    const int E = in_sizes[1] / 2;
    const int* row = ei;
    const int* col = ei + E;

    // workspace layout (floats)
    float* ws   = (float*)d_ws;
    size_t nd   = (size_t)N * D;
    float* A    = ws;                 // feature buffer
    float* B    = ws + nd;            // hw buffer
    float* C    = ws + 2 * nd;        // feature buffer
    float* dinv = ws + 3 * nd;
    float* deg  = dinv + N;
    float* nrm  = deg + N;            // E floats

    const int BS = 256;
    const int totND = N * D;
    const long long scatterT = (long long)(E + N) * 32;

    // ---- normalization precompute ----
    hipMemsetAsync(deg, 0, (size_t)N * sizeof(float), stream);
    k_degree<<<cdiv(E + N, BS), BS, 0, stream>>>(col, ew, deg, E, N);
    k_dinv  <<<cdiv(N, BS),     BS, 0, stream>>>(deg, dinv, N);
    k_norm  <<<cdiv(E, BS),     BS, 0, stream>>>(row, col, ew, dinv, nrm, E);

    const int gemmBlocks = cdiv(N, 32);

    // ---- layer 0: h1 = relu(scatter(x @ W_in) + b_in) -> C ----
    k_gemm64_wmma<<<gemmBlocks, BS, 0, stream>>>(x, W_in, B, N);
    k_init_bias  <<<cdiv(totND, BS), BS, 0, stream>>>(C, b_in, nullptr, totND);
    k_scatter64  <<<cdiv(scatterT, BS), BS, 0, stream>>>(row, col, nrm, dinv, B, C, E, N);
    k_relu       <<<cdiv(totND, BS), BS, 0, stream>>>(C, totND);

    // ---- layer 1: h2 = relu(scatter(h1 @ W1) + b1 + h1) -> A ----
    k_gemm64_wmma<<<gemmBlocks, BS, 0, stream>>>(C, W1, B, N);
    k_init_bias  <<<cdiv(totND, BS), BS, 0, stream>>>(A, b1, C, totND);
    k_scatter64  <<<cdiv(scatterT, BS), BS, 0, stream>>>(row, col, nrm, dinv, B, A, E, N);
    k_relu       <<<cdiv(totND, BS), BS, 0, stream>>>(A, totND);

    // ---- layer 2: h3 = relu(scatter(h2 @ W2) + b2 + h2) -> C ----
    k_gemm64_wmma<<<gemmBlocks, BS, 0, stream>>>(A, W2, B, N);
    k_init_bias  <<<cdiv(totND, BS), BS, 0, stream>>>(C, b2, A, totND);
    k_scatter64  <<<cdiv(scatterT, BS), BS, 0, stream>>>(row, col, nrm, dinv, B, C, E, N);
    k_relu       <<<cdiv(totND, BS), BS, 0, stream>>>(C, totND);

    // ---- final: out = scatter(h3 @ Wf) + bf  (d_out: [N]) ----
    float* outf = (float*)d_out;
    k_dot64      <<<cdiv(N, BS), BS, 0, stream>>>(C, Wf, B, N);
    k_init_scalar<<<cdiv(N, BS), BS, 0, stream>>>(outf, bf, N);
    k_scatter1   <<<cdiv(E + N, BS), BS, 0, stream>>>(row, col, nrm, dinv, B, outf, E, N);
}